// Neighbor_Aggregation_50268297232462
// MI455X (gfx1250) — compile-verified
//
#include <hip/hip_runtime.h>

// Problem constants (from reference): B=4, N=50000, E=800000, H_SIZE=64
#define BB 4
#define NN 50000
#define EE 800000
#define HS 64
#define WAVES_PER_BLOCK 8
#define EDGES_PER_WAVE 32
#define EDGES_PER_BLOCK (WAVES_PER_BLOCK * EDGES_PER_WAVE)  // 256

// ---------------------------------------------------------------------------
// Edge aggregation: one wave handles 32 edges.
//  - lane l loads edge (e0+l)'s metadata with coalesced, non-temporal loads
//  - per edge: v_readlane broadcasts (src,dst,w) to SGPRs
//  - 32 lanes load the 64-float H row as float2 (one 256B transaction,
//    saddr-form since the row base is wave-uniform)
//  - scaled values scatter via non-returning, saddr-form global_atomic_add_f32
//    (voffset = lane*8 is loop-invariant; scalar base = 1 SALU add per edge)
// ---------------------------------------------------------------------------
__global__ __launch_bounds__(256) void neighbor_agg_kernel(
    const float* __restrict__ H,        // [B, N, HS]
    const float* __restrict__ edge_w,   // [B, E]
    const int*   __restrict__ edge_src, // [B, E]
    const int*   __restrict__ edge_dst, // [B, E]
    float*       __restrict__ out)      // [B, N, HS]
{
    const int b    = blockIdx.y;
    const int lane = threadIdx.x & 31;
    const int wave = threadIdx.x >> 5;

    const float* Hb = H   + (size_t)b * NN * HS;
    float*       Ob = out + (size_t)b * NN * HS;
    const float* wb = edge_w   + (size_t)b * EE;
    const int*   sb = edge_src + (size_t)b * EE;
    const int*   db = edge_dst + (size_t)b * EE;

    const long long e0 = (long long)blockIdx.x * EDGES_PER_BLOCK
                       + (long long)wave * EDGES_PER_WAVE;
    const long long e  = e0 + lane;

    // Per-lane edge metadata (coalesced). Non-temporal: streamed exactly once,
    // keep L2 capacity for the H / out working set.
    int   my_s = 0, my_d = 0;
    float my_w = 0.0f;
    if (e < EE) {
        my_s = __builtin_nontemporal_load(sb + e);
        my_d = __builtin_nontemporal_load(db + e);
        my_w = __builtin_nontemporal_load(wb + e);
    }

    // Prefetch the 32 gathered H rows (256B each -> two 128B lines) so the
    // random-access latency overlaps the whole unrolled edge loop.
    {
        const char* pf = (const char*)(Hb + (size_t)my_d * HS);
        __builtin_prefetch(pf,       0, 3);   // global_prefetch_b8
        __builtin_prefetch(pf + 128, 0, 3);
    }

    // Loop-invariant per-lane byte offset into a 64-float row (float2/lane).
    const unsigned voff = (unsigned)lane * 8u;

#pragma unroll
    for (int j = 0; j < EDGES_PER_WAVE; ++j) {
        // Broadcast edge j's metadata to the whole wave (v_readlane -> SGPR,
        // so both row base addresses become scalar).
        const int   s = __builtin_amdgcn_readlane(my_s, j);
        const int   d = __builtin_amdgcn_readlane(my_d, j);
        const float w = __int_as_float(
            __builtin_amdgcn_readlane(__float_as_int(my_w), j));

        // 32 lanes x float2 = full 64-float row (saddr-form global_load_b64).
        const float2 h = *(const float2*)(Hb + (size_t)d * HS + lane * 2);

        const float mx = w * h.x;
        const float my = w * h.y;

        // Non-returning fp32 atomic adds, saddr form: scalar row base +
        // loop-invariant 32-bit lane offset. Native L2 RMW, STOREcnt-tracked.
        const float* obase = Ob + (size_t)s * HS;   // wave-uniform -> SGPR pair
        asm volatile(
            "global_atomic_add_f32 %0, %1, %3\n\t"
            "global_atomic_add_f32 %0, %2, %3 offset:4"
            :
            : "v"(voff), "v"(mx), "v"(my), "s"(obase)
            : "memory");
    }
}

// ---------------------------------------------------------------------------
// Zero the output (harness poisons d_out with 0xAA before timing).
// ---------------------------------------------------------------------------
__global__ __launch_bounds__(256) void zero_f4_kernel(float4* __restrict__ out, int n4) {
    int i = blockIdx.x * blockDim.x + threadIdx.x;
    if (i < n4) out[i] = make_float4(0.f, 0.f, 0.f, 0.f);
}

// ---------------------------------------------------------------------------
// Launch
// ---------------------------------------------------------------------------
extern "C" void kernel_launch(void* const* d_in, const int* in_sizes, int n_in,
                              void* d_out, int out_size, void* d_ws, size_t ws_size,
                              hipStream_t stream) {
    const float* H        = (const float*)d_in[0];
    const float* edge_w   = (const float*)d_in[1];
    const int*   edge_src = (const int*)d_in[2];
    const int*   edge_dst = (const int*)d_in[3];
    float*       out      = (float*)d_out;

    // Zero output: out_size = B*N*HS elements, multiple of 4.
    int n4 = out_size / 4;
    zero_f4_kernel<<<(n4 + 255) / 256, 256, 0, stream>>>((float4*)out, n4);

    dim3 grid((EE + EDGES_PER_BLOCK - 1) / EDGES_PER_BLOCK, BB);
    neighbor_agg_kernel<<<grid, WAVES_PER_BLOCK * 32, 0, stream>>>(
        H, edge_w, edge_src, edge_dst, out);
}